// Demo_ai_69329362092657
// MI455X (gfx1250) — compile-verified
//
#include <hip/hip_runtime.h>
#include <hip/hip_bf16.h>

#define NN   100000
#define NE   6400000
#define INF_ 128
#define HID  10
#define OUTF 3

typedef __attribute__((ext_vector_type(2))) float v2f;
typedef __attribute__((ext_vector_type(8))) float v8f;

// ---------------------------------------------------------------------------
// zero-fill (agg/deg accumulators must be re-zeroed every call)
// ---------------------------------------------------------------------------
__global__ __launch_bounds__(256) void zero_f32(float* __restrict__ p, int n) {
    int i = blockIdx.x * 256 + threadIdx.x;
    if (i < n) p[i] = 0.0f;
}

// ---------------------------------------------------------------------------
// H1 = relu(X @ W1 + b1)   via V_WMMA_F32_16X16X4_F32
// One wave32 computes a 16-row x 16-col tile (cols 10..15 padded with zeros).
// A layout: M = lane&15, K pair = {2*half, 2*half+1} across the 2 VGPRs.
// B layout: N = lane&15, same K pairing. C/D: M = v + 8*half, N = lane&15.
// ---------------------------------------------------------------------------
__global__ __launch_bounds__(256) void lin1_wmma(const float* __restrict__ X,
                                                 const float* __restrict__ W1,
                                                 const float* __restrict__ b1,
                                                 float* __restrict__ H,
                                                 int nTiles)
{
    __shared__ float Wp[INF_][16];               // W1 padded to 16 cols, 8KB
    int tid = threadIdx.x;
    for (int idx = tid; idx < INF_ * 16; idx += 256) {
        int k = idx >> 4, n = idx & 15;
        Wp[k][n] = (n < HID) ? W1[k * HID + n] : 0.0f;
    }
    __syncthreads();

    int wave = tid >> 5;
    int lane = tid & 31;
    int tile = blockIdx.x * 8 + wave;
    if (tile >= nTiles) return;                  // wave-uniform: EXEC all-1 inside

    int n    = lane & 15;                        // N for B/C/D, M for A
    int half = lane >> 4;
    int koff = half * 2;

    float bias = (n < HID) ? b1[n] : 0.0f;
    v8f c;
#pragma unroll
    for (int v = 0; v < 8; ++v) c[v] = bias;

    const float* xrow = X + (size_t)(tile * 16 + n) * INF_;
#pragma unroll 4
    for (int k = 0; k < INF_; k += 4) {
        v2f a, b;
        a.x = xrow[k + koff];
        a.y = xrow[k + koff + 1];
        b.x = Wp[k + koff][n];
        b.y = Wp[k + koff + 1][n];
        c = __builtin_amdgcn_wmma_f32_16x16x4_f32(false, a, false, b,
                                                  (short)0, c, false, false);
    }

    if (n < HID) {
        int mbase = tile * 16 + half * 8;
#pragma unroll
        for (int v = 0; v < 8; ++v) {
            float val = c[v];
            H[(size_t)(mbase + v) * HID + n] = val > 0.0f ? val : 0.0f;
        }
    }
}

// ---------------------------------------------------------------------------
// edge scatter: agg[dst] += H[src]; optionally deg[dst] += 1
// ---------------------------------------------------------------------------
__global__ __launch_bounds__(256) void edge_scatter(const long long* __restrict__ src,
                                                    const long long* __restrict__ dst,
                                                    const float* __restrict__ H,
                                                    float* __restrict__ agg,
                                                    float* __restrict__ deg,
                                                    int nE)
{
    int e = blockIdx.x * 256 + threadIdx.x;
    if (e >= nE) return;
    long long s = src[e];
    long long d = dst[e];
    const float* hp = H + (size_t)s * HID;
    float*       ap = agg + (size_t)d * HID;
#pragma unroll
    for (int j = 0; j < HID; ++j) atomicAdd(ap + j, hp[j]);
    if (deg) atomicAdd(deg + d, 1.0f);
}

// ---------------------------------------------------------------------------
// Hout = (agg/max(deg,1)) @ Wl + bl + Hin @ Wr           (per-node, K=10)
// ---------------------------------------------------------------------------
__global__ __launch_bounds__(256) void sage_post(const float* __restrict__ agg,
                                                 const float* __restrict__ deg,
                                                 const float* __restrict__ Hin,
                                                 const float* __restrict__ Wl,
                                                 const float* __restrict__ bl,
                                                 const float* __restrict__ Wr,
                                                 float* __restrict__ Hout,
                                                 int nNodes)
{
    int i = blockIdx.x * 256 + threadIdx.x;
    if (i >= nNodes) return;
    float d   = deg[i];
    float inv = 1.0f / (d < 1.0f ? 1.0f : d);
    float m[HID], h[HID], acc[HID];
#pragma unroll
    for (int j = 0; j < HID; ++j) {
        m[j]   = agg[(size_t)i * HID + j] * inv;
        h[j]   = Hin[(size_t)i * HID + j];
        acc[j] = bl[j];
    }
#pragma unroll
    for (int k = 0; k < HID; ++k) {
        float mk = m[k], hk = h[k];
#pragma unroll
        for (int j = 0; j < HID; ++j)
            acc[j] = fmaf(mk, Wl[k * HID + j], fmaf(hk, Wr[k * HID + j], acc[j]));
    }
#pragma unroll
    for (int j = 0; j < HID; ++j) Hout[(size_t)i * HID + j] = acc[j];
}

// ---------------------------------------------------------------------------
// final: conv2 post + lin2 fused:  out = (sage2(h)) @ W2 + b2   -> [N,3]
// ---------------------------------------------------------------------------
__global__ __launch_bounds__(256) void sage_post_lin2(const float* __restrict__ agg,
                                                      const float* __restrict__ deg,
                                                      const float* __restrict__ Hin,
                                                      const float* __restrict__ Wl,
                                                      const float* __restrict__ bl,
                                                      const float* __restrict__ Wr,
                                                      const float* __restrict__ W2,
                                                      const float* __restrict__ b2,
                                                      float* __restrict__ out,
                                                      int nNodes)
{
    int i = blockIdx.x * 256 + threadIdx.x;
    if (i >= nNodes) return;
    float d   = deg[i];
    float inv = 1.0f / (d < 1.0f ? 1.0f : d);
    float m[HID], h[HID], acc[HID];
#pragma unroll
    for (int j = 0; j < HID; ++j) {
        m[j]   = agg[(size_t)i * HID + j] * inv;
        h[j]   = Hin[(size_t)i * HID + j];
        acc[j] = bl[j];
    }
#pragma unroll
    for (int k = 0; k < HID; ++k) {
        float mk = m[k], hk = h[k];
#pragma unroll
        for (int j = 0; j < HID; ++j)
            acc[j] = fmaf(mk, Wl[k * HID + j], fmaf(hk, Wr[k * HID + j], acc[j]));
    }
    float o[OUTF];
#pragma unroll
    for (int c = 0; c < OUTF; ++c) o[c] = b2[c];
#pragma unroll
    for (int j = 0; j < HID; ++j) {
        float aj = acc[j];
#pragma unroll
        for (int c = 0; c < OUTF; ++c) o[c] = fmaf(aj, W2[j * OUTF + c], o[c]);
    }
#pragma unroll
    for (int c = 0; c < OUTF; ++c) out[(size_t)i * OUTF + c] = o[c];
}

// ---------------------------------------------------------------------------
extern "C" void kernel_launch(void* const* d_in, const int* in_sizes, int n_in,
                              void* d_out, int out_size, void* d_ws, size_t ws_size,
                              hipStream_t stream) {
    const float*     x   = (const float*)d_in[0];
    const long long* ei  = (const long long*)d_in[1];   // int64 [2, NE]
    const float*     W1  = (const float*)d_in[2];
    const float*     b1  = (const float*)d_in[3];
    const float*     Wl1 = (const float*)d_in[4];
    const float*     bl1 = (const float*)d_in[5];
    const float*     Wr1 = (const float*)d_in[6];
    const float*     Wl2 = (const float*)d_in[7];
    const float*     bl2 = (const float*)d_in[8];
    const float*     Wr2 = (const float*)d_in[9];
    const float*     W2  = (const float*)d_in[10];
    const float*     b2  = (const float*)d_in[11];
    float*           out = (float*)d_out;

    const long long* src = ei;
    const long long* dst = ei + NE;

    // workspace layout (bytes): H1 [0, 4e6), H2 [4e6, 8e6), agg [8e6, 12e6), deg [12e6, 12.4e6)
    char*  ws  = (char*)d_ws;
    float* H1  = (float*)(ws);
    float* H2  = (float*)(ws + 4000000);
    float* agg = (float*)(ws + 8000000);
    float* deg = (float*)(ws + 12000000);

    const int nTiles = NN / 16;                       // 6250 exactly
    const int linBlocks   = (nTiles + 7) / 8;         // 8 waves/block
    const int edgeBlocks  = (NE + 255) / 256;
    const int nodeBlocks  = (NN + 255) / 256;
    const int zeroAllN    = NN * HID + NN;            // agg + deg contiguous
    const int zeroAggN    = NN * HID;

    // 1. zero agg + deg
    zero_f32<<<(zeroAllN + 255) / 256, 256, 0, stream>>>(agg, zeroAllN);
    // 2. H1 = relu(x @ W1 + b1)   [WMMA f32]
    lin1_wmma<<<linBlocks, 256, 0, stream>>>(x, W1, b1, H1, nTiles);
    // 3. conv1 scatter (+ degree)
    edge_scatter<<<edgeBlocks, 256, 0, stream>>>(src, dst, H1, agg, deg, NE);
    // 4. conv1 combine
    sage_post<<<nodeBlocks, 256, 0, stream>>>(agg, deg, H1, Wl1, bl1, Wr1, H2, NN);
    // 5. re-zero agg
    zero_f32<<<(zeroAggN + 255) / 256, 256, 0, stream>>>(agg, zeroAggN);
    // 6. conv2 scatter (deg reused)
    edge_scatter<<<edgeBlocks, 256, 0, stream>>>(src, dst, H2, agg, nullptr, NE);
    // 7. conv2 combine + lin2 fused
    sage_post_lin2<<<nodeBlocks, 256, 0, stream>>>(agg, deg, H2, Wl2, bl2, Wr2, W2, b2, out, NN);
}